// EnhancedPEERLayer_6751688589561
// MI455X (gfx1250) — compile-verified
//
#include <hip/hip_runtime.h>
#include <hip/hip_bf16.h>
#include <math.h>

#define D_MODEL 1024
#define N_HEADS 8
#define D_KEY   128
#define TOP_K   16
#define N_SUB   128
#define SEQ     512
#define BATCH   2
#define TOKENS  (BATCH*SEQ)   // 1024

typedef __attribute__((ext_vector_type(2))) float v2f;
typedef __attribute__((ext_vector_type(8))) float v8f;

__device__ __forceinline__ v8f wmma4(v2f a, v2f b, v8f c) {
  // V_WMMA_F32_16X16X4_F32 : D = A(16x4) * B(4x16) + C(16x16), fp32
  return __builtin_amdgcn_wmma_f32_16x16x4_f32(
      /*neg_a=*/false, a, /*neg_b=*/false, b,
      /*c_mod=*/(short)0, c, /*reuse_a=*/false, /*reuse_b=*/false);
}

// ---------------------------------------------------------------------------
// Generic batched GEMM via WMMA fp32, register-blocked.
//   C[m,n] = sum_k A[m,k] * B(k,n) + bias[n]   (epilogue optional batchnorm)
// BT=1: B stored as N x K row-major (B[n*ldb+k])   -- "NT" gemm
// BT=0: B stored as K x N row-major (B[k*ldb+n])   -- "NN" gemm
// Wave tile = (16*MT) x (16*NT); block = 128 threads (4 waves) side by side
// in N. Per K-step of 4: MT A-fragments + NT B-fragments feed MT*NT WMMAs.
// ---------------------------------------------------------------------------
template<int BT, int EPI, int MT, int NT>
__global__ void gemm_wmma_kernel(
    const float* __restrict__ A, int lda, long sAb, long sAh,
    const float* __restrict__ B, int ldb, long sBb, long sBh,
    float*       __restrict__ C, int ldc, long sCb, long sCh,
    const float* __restrict__ bias, int K, int nh,
    const float* __restrict__ bn_w, const float* __restrict__ bn_b,
    const float* __restrict__ bn_mean, const float* __restrict__ bn_var)
{
  const int z  = blockIdx.z;
  const int zb = z / nh, zh = z % nh;
  A += (long)zb * sAb + (long)zh * sAh;
  B += (long)zb * sBb + (long)zh * sBh;
  C += (long)zb * sCb + (long)zh * sCh;

  const int lane = threadIdx.x & 31;
  const int wave = threadIdx.x >> 5;
  const int row0 = blockIdx.y * (16 * MT);
  const int col0 = (blockIdx.x * 4 + wave) * (16 * NT);
  const int m    = lane & 15;   // row/col within a 16x16 tile
  const int half = lane >> 4;   // lane group selects K pair {0,1} vs {2,3}

  v8f acc[MT][NT];
  #pragma unroll
  for (int mi = 0; mi < MT; ++mi)
    #pragma unroll
    for (int j = 0; j < NT; ++j) acc[mi][j] = (v8f){};

  const float* Arow[MT];
  #pragma unroll
  for (int mi = 0; mi < MT; ++mi)
    Arow[mi] = A + (long)(row0 + mi * 16 + m) * lda;

  const float* Brow[NT];
  if (BT) {
    #pragma unroll
    for (int j = 0; j < NT; ++j)
      Brow[j] = B + (long)(col0 + j * 16 + m) * ldb;
  }

  #pragma unroll 2
  for (int k0 = 0; k0 < K; k0 += 4) {
    const int ka = k0 + 2 * half;
    v2f a[MT];
    #pragma unroll
    for (int mi = 0; mi < MT; ++mi) { a[mi].x = Arow[mi][ka]; a[mi].y = Arow[mi][ka + 1]; }
    #pragma unroll
    for (int j = 0; j < NT; ++j) {
      v2f b;
      if (BT) {
        b.x = Brow[j][ka]; b.y = Brow[j][ka + 1];
      } else {
        b.x = B[(long)ka * ldb + col0 + j * 16 + m];
        b.y = B[(long)(ka + 1) * ldb + col0 + j * 16 + m];
      }
      #pragma unroll
      for (int mi = 0; mi < MT; ++mi) acc[mi][j] = wmma4(a[mi], b, acc[mi][j]);
    }
  }

  #pragma unroll
  for (int j = 0; j < NT; ++j) {
    const int n = col0 + j * 16 + m;
    float bia = bias ? bias[n] : 0.0f;
    float scl = 1.0f, sft = 0.0f;
    if (EPI == 1) { // inference batchnorm: (v - mean)*rsqrt(var+eps)*w + b
      scl = bn_w[n] * rsqrtf(bn_var[n] + 1e-5f);
      sft = bn_b[n] - bn_mean[n] * scl;
    }
    #pragma unroll
    for (int mi = 0; mi < MT; ++mi) {
      #pragma unroll
      for (int i = 0; i < 8; ++i) {
        const int r = row0 + mi * 16 + i + 8 * half;  // C map: VGPR i -> rows i, i+8
        float v = acc[mi][j][i] + bia;
        if (EPI == 1) v = v * scl + sft;
        C[(long)r * ldc + n] = v;
      }
    }
  }
}

// ---------------------------------------------------------------------------
// PEER: per (token, head) -> subkey scores, top-16 product-key retrieval,
// expert similarity softmax, weighted expert combine.
// ---------------------------------------------------------------------------
__global__ void peer_kernel(const float* __restrict__ q,
                            const float* __restrict__ sub_keys,
                            const float* __restrict__ expert_weights,
                            float* __restrict__ peer)
{
  const int th  = blockIdx.x;          // 0 .. TOKENS*N_HEADS-1
  const int tok = th >> 3, h = th & 7;
  const int t   = threadIdx.x;         // 256 threads

  __shared__ float qloc[D_KEY];
  __shared__ float s1[N_SUB], s2[N_SUB];
  __shared__ float red[256];
  __shared__ int   redi[256];
  __shared__ int   chosen[TOP_K];
  __shared__ float rw[TOP_K];

  if (t < D_KEY) qloc[t] = q[(long)tok * D_MODEL + h * D_KEY + t];
  __syncthreads();

  { // sub-key scores: s1[n] = q1 . K1[n], s2[n] = q2 . K2[n]
    const int which = t >> 7;          // 0 -> s1, 1 -> s2
    const int n = t & 127;
    const float* kr = sub_keys + ((long)which * N_SUB + n) * (D_KEY / 2);
    const float* qh = qloc + which * (D_KEY / 2);
    float acc = 0.f;
    #pragma unroll 8
    for (int c = 0; c < D_KEY / 2; ++c) acc += qh[c] * kr[c];
    if (which == 0) s1[n] = acc; else s2[n] = acc;
  }
  __syncthreads();

  // top-16 of combined score s1[i] + s2[j]  (index c = i*128 + j)
  for (int p = 0; p < TOP_K; ++p) {
    float bv = -INFINITY; int bi = 0x7fffffff;
    for (int c = t * 64; c < t * 64 + 64; ++c) {
      const int i = c >> 7, j = c & 127;
      const float v = s1[i] + s2[j];
      bool taken = false;
      for (int e = 0; e < p; ++e) taken |= (chosen[e] == c);
      if (!taken && (v > bv || (v == bv && c < bi))) { bv = v; bi = c; }
    }
    red[t] = bv; redi[t] = bi; __syncthreads();
    for (int s = 128; s > 0; s >>= 1) {
      if (t < s) {
        const float ov = red[t + s]; const int oi = redi[t + s];
        if (ov > red[t] || (ov == red[t] && oi < redi[t])) { red[t] = ov; redi[t] = oi; }
      }
      __syncthreads();
    }
    if (t == 0) chosen[p] = redi[0];
    __syncthreads();
  }

  // sims: wave w computes experts w and w+8 (dot length 128, wave32 reduce)
  {
    const int lane = t & 31, w = t >> 5;
    for (int kk = w; kk < TOP_K; kk += 8) {
      const float* er = expert_weights + (long)chosen[kk] * D_KEY;
      float acc = 0.f;
      #pragma unroll
      for (int d = lane; d < D_KEY; d += 32) acc += qloc[d] * er[d];
      #pragma unroll
      for (int off = 16; off > 0; off >>= 1) acc += __shfl_xor(acc, off, 32);
      if (lane == 0) red[kk] = acc;
    }
  }
  __syncthreads();
  if (t == 0) { // softmax over 16 sims
    float mx = -INFINITY;
    for (int k = 0; k < TOP_K; ++k) mx = fmaxf(mx, red[k]);
    float s = 0.f;
    for (int k = 0; k < TOP_K; ++k) { const float e = __expf(red[k] - mx); rw[k] = e; s += e; }
    const float inv = 1.f / s;
    for (int k = 0; k < TOP_K; ++k) rw[k] *= inv;
  }
  __syncthreads();
  if (t < D_KEY) { // weighted combine of gathered experts
    float acc = 0.f;
    #pragma unroll
    for (int k = 0; k < TOP_K; ++k)
      acc += rw[k] * expert_weights[(long)chosen[k] * D_KEY + t];
    peer[(long)tok * D_MODEL + h * D_KEY + t] = acc;
  }
}

// ---------------------------------------------------------------------------
// Row softmax over attention scores (row length SEQ=512), scale 1/sqrt(128).
// ---------------------------------------------------------------------------
__global__ void attn_softmax_kernel(float* __restrict__ att)
{
  const long row = blockIdx.x;             // BATCH*N_HEADS*SEQ rows
  float* p = att + row * SEQ;
  const int t = threadIdx.x;               // 256 threads, 2 elems each
  __shared__ float red[256];
  const float sc = 0.08838834764831845f;   // 1/sqrt(128)
  float v0 = p[t] * sc, v1 = p[t + 256] * sc;
  red[t] = fmaxf(v0, v1); __syncthreads();
  for (int s = 128; s > 0; s >>= 1) { if (t < s) red[t] = fmaxf(red[t], red[t + s]); __syncthreads(); }
  const float mx = red[0]; __syncthreads();
  v0 = __expf(v0 - mx); v1 = __expf(v1 - mx);
  red[t] = v0 + v1; __syncthreads();
  for (int s = 128; s > 0; s >>= 1) { if (t < s) red[t] += red[t + s]; __syncthreads(); }
  const float inv = 1.0f / red[0];
  p[t] = v0 * inv; p[t + 256] = v1 * inv;
}

// attn_weights = mean over heads of att
__global__ void attn_mean_kernel(const float* __restrict__ att, float* __restrict__ aw)
{
  const long i  = (long)blockIdx.x * blockDim.x + threadIdx.x;  // BATCH*SEQ*SEQ
  const long bb = i / ((long)SEQ * SEQ);
  const long qk = i % ((long)SEQ * SEQ);
  const float* base = att + bb * (long)N_HEADS * SEQ * SEQ + qk;
  float acc = 0.f;
  #pragma unroll
  for (int h = 0; h < N_HEADS; ++h) acc += base[(long)h * SEQ * SEQ];
  aw[i] = acc * (1.0f / N_HEADS);
}

// y = rmsnorm(x + peer_proj + ao_proj) * rms_w
__global__ void final_kernel(const float* __restrict__ x,
                             const float* __restrict__ peer_proj,
                             const float* __restrict__ ao_proj,
                             const float* __restrict__ rms_w,
                             float* __restrict__ y)
{
  const long row = blockIdx.x; const int t = threadIdx.x; // 256 thr, 4 cols each
  __shared__ float red[256];
  const float* xr = x + row * D_MODEL;
  const float* pr = peer_proj + row * D_MODEL;
  const float* ar = ao_proj + row * D_MODEL;
  float v[4]; float ss = 0.f;
  #pragma unroll
  for (int i = 0; i < 4; ++i) {
    const int c = t + 256 * i;
    v[i] = xr[c] + pr[c] + ar[c];
    ss += v[i] * v[i];
  }
  red[t] = ss; __syncthreads();
  for (int s = 128; s > 0; s >>= 1) { if (t < s) red[t] += red[t + s]; __syncthreads(); }
  const float inv = rsqrtf(red[0] / (float)D_MODEL + 1e-6f);
  #pragma unroll
  for (int i = 0; i < 4; ++i) {
    const int c = t + 256 * i;
    y[row * D_MODEL + c] = v[i] * inv * rms_w[c];
  }
}

// ---------------------------------------------------------------------------
extern "C" void kernel_launch(void* const* d_in, const int* in_sizes, int n_in,
                              void* d_out, int out_size, void* d_ws, size_t ws_size,
                              hipStream_t stream) {
  (void)in_sizes; (void)n_in; (void)out_size; (void)ws_size;
  const float* x        = (const float*)d_in[0];
  const float* wq       = (const float*)d_in[1];
  const float* bq       = (const float*)d_in[2];
  const float* bn_w     = (const float*)d_in[3];
  const float* bn_b     = (const float*)d_in[4];
  const float* bn_mean  = (const float*)d_in[5];
  const float* bn_var   = (const float*)d_in[6];
  const float* sub_keys = (const float*)d_in[7];
  const float* expert_w = (const float*)d_in[8];
  const float* wo       = (const float*)d_in[9];
  const float* bo       = (const float*)d_in[10];
  const float* inpw     = (const float*)d_in[11];
  const float* inpb     = (const float*)d_in[12];
  const float* attn_ow  = (const float*)d_in[13];
  const float* attn_ob  = (const float*)d_in[14];
  const float* rms_w    = (const float*)d_in[15];

  float* out_y  = (float*)d_out;                          // 1024*1024
  float* out_aw = (float*)d_out + (long)TOKENS * D_MODEL; // 2*512*512

  // workspace layout (floats)
  float* ws        = (float*)d_ws;
  float* q         = ws;                       // 1M (reused as ao later)
  float* qkv       = ws + (1L << 20);          // 3M
  float* peer      = ws + (4L << 20);          // 1M
  float* peer_proj = ws + (5L << 20);          // 1M
  float* att       = ws + (6L << 20);          // 4M
  float* ao        = q;                        // reuse: q dead after peer
  float* ao_proj   = qkv;                      // reuse: qkv dead after AV gemm

  const dim3 B128(128), B256(256);

  // 1) q = BN(x @ wq^T + bq)           [1024 x 1024 x 1024]
  gemm_wmma_kernel<1,1,2,4><<<dim3(D_MODEL/256, TOKENS/32, 1), B128, 0, stream>>>(
      x, D_MODEL, 0, 0, wq, D_MODEL, 0, 0, q, D_MODEL, 0, 0,
      bq, D_MODEL, 1, bn_w, bn_b, bn_mean, bn_var);

  // 2) qkv = x @ in_proj_w^T + b       [1024 x 3072 x 1024]
  gemm_wmma_kernel<1,0,2,4><<<dim3(3*D_MODEL/256, TOKENS/32, 1), B128, 0, stream>>>(
      x, D_MODEL, 0, 0, inpw, D_MODEL, 0, 0, qkv, 3*D_MODEL, 0, 0,
      inpb, D_MODEL, 1, nullptr, nullptr, nullptr, nullptr);

  // 3) PEER retrieval per (token, head)
  peer_kernel<<<dim3(TOKENS * N_HEADS), B256, 0, stream>>>(q, sub_keys, expert_w, peer);

  // 4) peer_proj = peer @ wo^T + bo    [1024 x 1024 x 1024]
  gemm_wmma_kernel<1,0,2,4><<<dim3(D_MODEL/256, TOKENS/32, 1), B128, 0, stream>>>(
      peer, D_MODEL, 0, 0, wo, D_MODEL, 0, 0, peer_proj, D_MODEL, 0, 0,
      bo, D_MODEL, 1, nullptr, nullptr, nullptr, nullptr);

  // 5) attention scores: att[b,h] = qa @ ka^T     [512 x 512 x 128] x16
  gemm_wmma_kernel<1,0,2,4><<<dim3(SEQ/256, SEQ/32, BATCH*N_HEADS), B128, 0, stream>>>(
      qkv,            3*D_MODEL, (long)SEQ*3*D_MODEL, D_MODEL/N_HEADS,   // qa
      qkv + D_MODEL,  3*D_MODEL, (long)SEQ*3*D_MODEL, D_MODEL/N_HEADS,   // ka
      att,            SEQ, (long)N_HEADS*SEQ*SEQ, (long)SEQ*SEQ,
      nullptr, D_MODEL/N_HEADS, N_HEADS, nullptr, nullptr, nullptr, nullptr);

  // 6) softmax rows (with 1/sqrt(128) scale)
  attn_softmax_kernel<<<dim3(BATCH*N_HEADS*SEQ), B256, 0, stream>>>(att);

  // 7) ao[b,h] = att @ va              [512 x 128 x 512] x16 (NN gemm)
  gemm_wmma_kernel<0,0,2,2><<<dim3(1, SEQ/32, BATCH*N_HEADS), B128, 0, stream>>>(
      att,             SEQ, (long)N_HEADS*SEQ*SEQ, (long)SEQ*SEQ,
      qkv + 2*D_MODEL, 3*D_MODEL, (long)SEQ*3*D_MODEL, D_MODEL/N_HEADS,  // va
      ao,              D_MODEL, (long)SEQ*D_MODEL, D_MODEL/N_HEADS,
      nullptr, SEQ, N_HEADS, nullptr, nullptr, nullptr, nullptr);

  // 8) ao_proj = ao @ attn_ow^T + attn_ob   [1024 x 1024 x 1024]
  gemm_wmma_kernel<1,0,2,4><<<dim3(D_MODEL/256, TOKENS/32, 1), B128, 0, stream>>>(
      ao, D_MODEL, 0, 0, attn_ow, D_MODEL, 0, 0, ao_proj, D_MODEL, 0, 0,
      attn_ob, D_MODEL, 1, nullptr, nullptr, nullptr, nullptr);

  // 9) attn_weights = mean_h att
  attn_mean_kernel<<<dim3((BATCH*SEQ*SEQ)/256), B256, 0, stream>>>(att, out_aw);

  // 10) y = rmsnorm(x + peer_proj + ao_proj) * rms_w
  final_kernel<<<dim3(TOKENS), B256, 0, stream>>>(x, peer_proj, ao_proj, rms_w, out_y);
}